// PixelPerfect_70274254897153
// MI455X (gfx1250) — compile-verified
//
#include <hip/hip_runtime.h>

#define B_DIM   8
#define C_DIM   256
#define HW      65536
#define K_VALS  3
#define N_FEAT  128
#define NBKT    4096
#define CAP     2048
#define DEPTH   8            // async pipeline depth (buffers per thread)

#ifndef __has_builtin
#define __has_builtin(x) 0
#endif

#if __has_builtin(__builtin_amdgcn_global_load_async_to_lds_b128)
#define USE_ASYNC_LDS 1
typedef int v4i_t __attribute__((vector_size(16)));
typedef __attribute__((address_space(1))) v4i_t gv4i_t;   // global (AS1)
typedef __attribute__((address_space(3))) v4i_t lv4i_t;   // LDS    (AS3)
#else
#define USE_ASYNC_LDS 0
#endif

// ---------------------------------------------------------------------------
// Kernel 1: maxv[b][p] = max_c x[b][c][p]
// 512 blocks x 256 threads; each thread owns 4 consecutive pixels (float4).
// CDNA5 path: depth-8 async global->LDS pipeline (s_wait_asynccnt), unrolled
// so all buffer indices / LDS offsets are compile-time constants.
// ---------------------------------------------------------------------------
__global__ __launch_bounds__(256) void k_channel_max(const float* __restrict__ x,
                                                     float* __restrict__ maxv) {
    const int tid = threadIdx.x;
    const int b   = blockIdx.x >> 6;          // 64 tiles per batch
    const int p0  = (blockIdx.x & 63) << 10;  // 1024 pixels per block
    const float* gbase = x + (size_t)b * C_DIM * HW + p0 + tid * 4;

    float4 m;
    m.x = m.y = m.z = m.w = -3.402823466e+38f;

#if USE_ASYNC_LDS
    __shared__ float4 lbuf[DEPTH][256];       // 32 KB: one slot per thread per buffer

    // prologue: fill the pipeline with channels 0..DEPTH-1
    #pragma unroll
    for (int j = 0; j < DEPTH; ++j) {
        __builtin_amdgcn_global_load_async_to_lds_b128(
            (gv4i_t*)(gbase + (size_t)j * HW),
            (lv4i_t*)&lbuf[j][tid], 0, 0);
    }

    // steady state: consume channel cc+j from buffer j, re-issue channel cc+j+DEPTH
    for (int cc = 0; cc < C_DIM - DEPTH; cc += DEPTH) {
        #pragma unroll
        for (int j = 0; j < DEPTH; ++j) {
            asm volatile("s_wait_asynccnt 7" ::: "memory");   // oldest (cc+j) landed
            float4 v = lbuf[j][tid];
            m.x = fmaxf(m.x, v.x);
            m.y = fmaxf(m.y, v.y);
            m.z = fmaxf(m.z, v.z);
            m.w = fmaxf(m.w, v.w);
            __builtin_amdgcn_global_load_async_to_lds_b128(
                (gv4i_t*)(gbase + (size_t)(cc + DEPTH + j) * HW),
                (lv4i_t*)&lbuf[j][tid], 0, 0);
        }
    }

    // tail: drain the last DEPTH channels
    asm volatile("s_wait_asynccnt 0" ::: "memory");
    #pragma unroll
    for (int j = 0; j < DEPTH; ++j) {
        float4 v = lbuf[j][tid];
        m.x = fmaxf(m.x, v.x);
        m.y = fmaxf(m.y, v.y);
        m.z = fmaxf(m.z, v.z);
        m.w = fmaxf(m.w, v.w);
    }
#else
    #pragma unroll 4
    for (int c = 0; c < C_DIM; ++c) {
        float4 v = *(const float4*)(gbase + (size_t)c * HW);
        m.x = fmaxf(m.x, v.x);
        m.y = fmaxf(m.y, v.y);
        m.z = fmaxf(m.z, v.z);
        m.w = fmaxf(m.w, v.w);
    }
#endif
    *(float4*)(maxv + (size_t)b * HW + p0 + tid * 4) = m;
}

// ---------------------------------------------------------------------------
// Kernel 2: per batch (1 block, 1024 thr): radix-select top-128 pixels of
// maxv, exact rank-order them, recompute per-pixel channel top-3, write out.
// Output layout (flat, return order): idxk[8][3][128] | topk[8][3][128] |
// idxn[8][1][128], all as float.
// ---------------------------------------------------------------------------
__global__ __launch_bounds__(1024) void k_select_gather(const float* __restrict__ x,
                                                        const float* __restrict__ maxv,
                                                        float* __restrict__ out) {
    const int b   = blockIdx.x;
    const int tid = threadIdx.x;

    __shared__ int      hist[NBKT];     // 16 KB
    __shared__ int      groupSum[64];
    __shared__ unsigned cVal[CAP];      // 8 KB
    __shared__ int      cIdx[CAP];      // 8 KB
    __shared__ int      selIdx[N_FEAT];
    __shared__ int      sh_T;
    __shared__ int      sh_m;

    for (int i = tid; i < NBKT; i += 1024) hist[i] = 0;
    if (tid == 0) sh_m = 0;
    if (tid < N_FEAT) selIdx[tid] = 0;
    __syncthreads();

    const float* mb = maxv + (size_t)b * HW;

    // ---- histogram of top-12 bits of the order-preserving uint transform ----
    for (int j = 0; j < HW / 1024; ++j) {
        int i = tid + j * 1024;
        int bits = __float_as_int(mb[i]);
        unsigned u = (unsigned)bits ^ (unsigned)((bits >> 31) | 0x80000000);
        atomicAdd(&hist[u >> 20], 1);
    }
    __syncthreads();

    // ---- two-level reverse scan to find threshold bucket T ----
    if (tid < 64) {
        int s = 0;
        for (int i = 0; i < 64; ++i) s += hist[(tid << 6) + i];
        groupSum[tid] = s;
    }
    __syncthreads();
    if (tid == 0) {
        int acc = 0, T = 0;
        for (int g = 63; g >= 0; --g) {
            int gs = groupSum[g];
            if (acc + gs >= N_FEAT) {
                for (int bkt = (g << 6) + 63; bkt >= (g << 6); --bkt) {
                    int h = hist[bkt];
                    if (acc + h >= N_FEAT) { T = bkt; break; }
                    acc += h;
                }
                break;
            }
            acc += gs;
        }
        sh_T = T;
    }
    __syncthreads();
    const unsigned T = (unsigned)sh_T;

    // ---- collect all candidates with bucket >= T ----
    for (int j = 0; j < HW / 1024; ++j) {
        int i = tid + j * 1024;
        int bits = __float_as_int(mb[i]);
        unsigned u = (unsigned)bits ^ (unsigned)((bits >> 31) | 0x80000000);
        if ((u >> 20) >= T) {
            int pos = atomicAdd(&sh_m, 1);
            if (pos < CAP) { cVal[pos] = u; cIdx[pos] = i; }
        }
    }
    __syncthreads();

    // ---- exact rank among candidates: descending value, ties -> lower index ----
    int m = sh_m;
    if (m > CAP) m = CAP;
    if (tid < m) {
        unsigned u = cVal[tid];
        int idx = cIdx[tid];
        int r = 0;
        for (int j = 0; j < m; ++j) {
            unsigned uj = cVal[j];
            r += (int)(uj > u) + (int)((uj == u) && (cIdx[j] < idx));
        }
        if (r < N_FEAT) selIdx[r] = idx;
    }
    __syncthreads();

    // ---- recompute channel top-3 for the 128 selected pixels, write outputs ----
    if (tid < N_FEAT) {
        int p = selIdx[tid];
        if (p < 0) p = 0;
        if (p >= HW) p = HW - 1;
        const float* xb = x + (size_t)b * C_DIM * HW + p;
        float v0 = -3.402823466e+38f, v1 = v0, v2 = v0;
        int i0 = 0, i1 = 0, i2 = 0;
        #pragma unroll 4
        for (int c = 0; c < C_DIM; ++c) {
            float v = xb[(size_t)c * HW];
            if (v > v0)      { v2 = v1; i2 = i1; v1 = v0; i1 = i0; v0 = v; i0 = c; }
            else if (v > v1) { v2 = v1; i2 = i1; v1 = v;  i1 = c; }
            else if (v > v2) { v2 = v;  i2 = c; }
        }
        float* idxk = out;                                    // [8][3][128]
        float* topk = out + B_DIM * K_VALS * N_FEAT;          // [8][3][128]
        float* idxn = out + 2 * B_DIM * K_VALS * N_FEAT;      // [8][1][128]
        int base = b * K_VALS * N_FEAT + tid;
        idxk[base + 0 * N_FEAT] = (float)i0;
        idxk[base + 1 * N_FEAT] = (float)i1;
        idxk[base + 2 * N_FEAT] = (float)i2;
        topk[base + 0 * N_FEAT] = v0;
        topk[base + 1 * N_FEAT] = v1;
        topk[base + 2 * N_FEAT] = v2;
        idxn[b * N_FEAT + tid]  = (float)p;
    }
}

extern "C" void kernel_launch(void* const* d_in, const int* in_sizes, int n_in,
                              void* d_out, int out_size, void* d_ws, size_t ws_size,
                              hipStream_t stream) {
    (void)in_sizes; (void)n_in; (void)out_size; (void)ws_size;
    const float* x   = (const float*)d_in[0];
    float* out       = (float*)d_out;
    float* maxv      = (float*)d_ws;              // 8 * 65536 floats = 2 MB scratch

    k_channel_max<<<dim3(B_DIM * 64), dim3(256), 0, stream>>>(x, maxv);
    k_select_gather<<<dim3(B_DIM), dim3(1024), 0, stream>>>(x, maxv, out);
}